// SpectralCompactnessLoss_65738769433115
// MI455X (gfx1250) — compile-verified
//
#include <hip/hip_runtime.h>

typedef float v2f __attribute__((ext_vector_type(2)));
typedef float v8f __attribute__((ext_vector_type(8)));

#define B_  2
#define C_  32
#define K_  2
#define N_  393216   // 64*64*96

// ws layout (floats):
//   [0,128)   centerSum[b][k][c]
//   [128,132) counts[b][k]
//   [132,136) weightedSum[b][k]
#define WS_CNT  128
#define WS_WSUM 132
#define WS_TOT  136

__global__ void zero_ws_kernel(float* ws) {
    int i = threadIdx.x;
    if (i < WS_TOT) ws[i] = 0.0f;
}

// Pass 1: per-class channel sums via fp32 WMMA:  D(16x16) = A(16x4 one-hot^T) x B(4x16 feat) + D
// A layout (32-bit 16x4): lane l<16 -> M=l, v0=K0, v1=K1 ; lane 16+l -> M=l, v0=K2, v1=K3
// B layout (4x16):        lane l<16 -> v0=row0[l], v1=row1[l] ; lane 16+l -> v0=row2[l], v1=row3[l]
// Each wave owns a CONTIGUOUS pixel chunk; per lane the 4 pixels of an iteration are one
// float4 (global_load_b128), so every byte of every fetched cacheline is consumed.
__global__ __launch_bounds__(256) void centers_wmma_kernel(
        const float* __restrict__ feat, const int* __restrict__ tgt,
        float* __restrict__ ws) {
    const int lane = threadIdx.x & 31;
    const int hi   = lane >> 4;          // 0: low half-wave, 1: high half-wave
    const int m    = lane & 15;          // row (class) for A, column (channel) for B
    const int wid  = blockIdx.x * (blockDim.x >> 5) + (threadIdx.x >> 5);
    const int nwaves = gridDim.x * (blockDim.x >> 5);
    // contiguous chunk per wave, multiple of 4
    const int chunk  = ((N_ + nwaves * 4 - 1) / (nwaves * 4)) * 4;
    const int nBeg   = wid * chunk;
    const int nEnd   = (nBeg + chunk < N_) ? (nBeg + chunk) : N_;

    for (int b = 0; b < B_; ++b) {
        v8f acc0 = {};   // channels 0..15
        v8f acc1 = {};   // channels 16..31
        int cnt0 = 0, npix = 0;
        const float* fb = feat + (size_t)b * C_ * N_;
        const int*   tb = tgt  + (size_t)b * N_;

        for (int n = nBeg; n + 3 < nEnd; n += 4) {
            const int4 tq = *(const int4*)(tb + n);          // targets of the 4 pixels

            // one-hot A tile: a.x covers pixel (hi?2:0), a.y covers pixel (hi?3:1)
            v2f a;
            a.x = ((hi ? tq.z : tq.x) == m) ? 1.0f : 0.0f;
            a.y = ((hi ? tq.w : tq.y) == m) ? 1.0f : 0.0f;

            // 16B-aligned b128 loads: 4 consecutive pixels of this lane's channel
            const float4 qlo = *(const float4*)(fb + (size_t)m        * N_ + n);
            const float4 qhi = *(const float4*)(fb + (size_t)(m + 16) * N_ + n);
            v2f b0, b1;
            b0.x = hi ? qlo.z : qlo.x;   // pixel jx = hi?2:0
            b0.y = hi ? qlo.w : qlo.y;   // pixel jy = hi?3:1
            b1.x = hi ? qhi.z : qhi.x;
            b1.y = hi ? qhi.w : qhi.y;

            acc0 = __builtin_amdgcn_wmma_f32_16x16x4_f32(
                       false, a, false, b0, (short)0, acc0, false, false);
            acc1 = __builtin_amdgcn_wmma_f32_16x16x4_f32(
                       false, a, false, b1, (short)0, acc1, false, false);

            cnt0 += (tq.x == 0) + (tq.y == 0) + (tq.z == 0) + (tq.w == 0);
            npix += 4;
        }

        // D rows: VGPR r, lanes 0-15 -> M=r, N=lane.  Row 0 = class 0 sums, row 1 = class 1 sums.
        if (lane < 16) {
            atomicAdd(&ws[(b * K_ + 0) * C_ + m],      acc0[0]);
            atomicAdd(&ws[(b * K_ + 0) * C_ + 16 + m], acc1[0]);
            atomicAdd(&ws[(b * K_ + 1) * C_ + m],      acc0[1]);
            atomicAdd(&ws[(b * K_ + 1) * C_ + 16 + m], acc1[1]);
        }
        if (lane == 0 && npix > 0) {
            atomicAdd(&ws[WS_CNT + b * K_ + 0], (float)cnt0);
            atomicAdd(&ws[WS_CNT + b * K_ + 1], (float)(npix - cnt0));
        }
    }
}

// Pass 2: per-pixel confidence-weighted L2 distance to own-class center, reduced per (b,class).
__global__ __launch_bounds__(256) void dist_loss_kernel(
        const float* __restrict__ feat, const float* __restrict__ pred,
        const int* __restrict__ tgt, float* __restrict__ ws) {
    __shared__ float centers[B_ * K_ * C_];   // 128 floats
    __shared__ float accs[B_ * K_];
    const int tid = threadIdx.x;
    if (tid < B_ * K_ * C_) {
        const int b = tid >> 6, k = (tid >> 5) & 1;
        const float cnt = ws[WS_CNT + b * K_ + k];
        centers[tid] = ws[tid] / fmaxf(cnt, 1.0f);
    }
    if (tid < B_ * K_) accs[tid] = 0.0f;
    __syncthreads();

    float lacc[B_ * K_] = {0.0f, 0.0f, 0.0f, 0.0f};
    const long long total  = (long long)B_ * N_;
    const long long stride = (long long)gridDim.x * blockDim.x;
    for (long long p = (long long)blockIdx.x * blockDim.x + tid; p < total; p += stride) {
        const int b = (int)(p / N_);
        const int n = (int)(p - (long long)b * N_);
        const int t = tgt[p];

        // prefetch next grid-stride iteration's lines (global_prefetch_b8)
        if (p + stride < total) {
            __builtin_prefetch(&tgt[p + stride], 0, 1);
            __builtin_prefetch(&pred[p + stride], 0, 1);
        }

        const float p0 = pred[((size_t)b * K_ + 0) * N_ + n];
        const float p1 = pred[((size_t)b * K_ + 1) * N_ + n];
        const float mx = fmaxf(p0, p1);
        const float e0 = __expf(p0 - mx), e1 = __expf(p1 - mx);
        const float conf = (t ? e1 : e0) / (e0 + e1);

        const float* cen = &centers[(b * K_ + t) * C_];
        float d2 = 0.0f;
        #pragma unroll
        for (int c = 0; c < C_; ++c) {
            const float f  = feat[((size_t)b * C_ + c) * N_ + n];  // coalesced across lanes
            const float df = f - cen[c];
            d2 = fmaf(df, df, d2);
        }
        lacc[b * K_ + t] += sqrtf(d2) * conf;
    }

    atomicAdd(&accs[0], lacc[0]);   // ds_add_f32 LDS atomics
    atomicAdd(&accs[1], lacc[1]);
    atomicAdd(&accs[2], lacc[2]);
    atomicAdd(&accs[3], lacc[3]);
    __syncthreads();
    if (tid < B_ * K_) atomicAdd(&ws[WS_WSUM + tid], accs[tid]);
}

__global__ void finalize_kernel(const float* __restrict__ ws, float* __restrict__ out) {
    if (threadIdx.x == 0) {
        float total = 0.0f, nvalid = 0.0f;
        for (int i = 0; i < B_ * K_; ++i) {
            const float npix  = ws[WS_CNT + i];
            const float cl    = ws[WS_WSUM + i] / fmaxf(npix, 1.0f);
            const float valid = (npix >= 2.0f) ? 1.0f : 0.0f;
            total  += cl * valid;
            nvalid += valid;
        }
        out[0] = (nvalid > 0.0f) ? (total / nvalid) : 0.0f;
    }
}

extern "C" void kernel_launch(void* const* d_in, const int* in_sizes, int n_in,
                              void* d_out, int out_size, void* d_ws, size_t ws_size,
                              hipStream_t stream) {
    (void)in_sizes; (void)n_in; (void)out_size; (void)ws_size;
    const float* feat = (const float*)d_in[0];
    const float* pred = (const float*)d_in[1];
    const int*   tgt  = (const int*)d_in[2];
    float* ws  = (float*)d_ws;
    float* out = (float*)d_out;

    zero_ws_kernel<<<1, 256, 0, stream>>>(ws);
    centers_wmma_kernel<<<512, 256, 0, stream>>>(feat, tgt, ws);
    dist_loss_kernel<<<1024, 256, 0, stream>>>(feat, pred, tgt, ws);
    finalize_kernel<<<1, 32, 0, stream>>>(ws, out);
}